// RoiPoolingConv_18442589569936
// MI455X (gfx1250) — compile-verified
//
#include <hip/hip_runtime.h>
#include <math.h>

typedef __attribute__((ext_vector_type(2))) float v2f;
typedef __attribute__((ext_vector_type(4))) float v4f;
typedef __attribute__((ext_vector_type(8))) float v8f;

#define POOL   7
#define NROI   512
#define W_IMG  128
#define C_IMG  1024
#define NSLOT  14            // 2*POOL row/col slots ({i0} ++ {i1})
#define HPAD   17            // LDS channel pad (bank-conflict avoidance)
#define WAVES  8
#define CTILES (C_IMG / 16)  // 64 channel tiles of 16

__device__ __forceinline__ int imin(int a, int b) { return a < b ? a : b; }

// D = A(16x4 f32) x B(4x16 f32) + C(16x16 f32), wave32 WMMA.
__device__ __forceinline__ v8f wmma_f32_k4(v2f a, v2f b, v8f c) {
    return __builtin_amdgcn_wmma_f32_16x16x4_f32(
        /*neg_a=*/false, a, /*neg_b=*/false, b,
        /*c_mod=*/(short)0, c, /*reuse_a=*/false, /*reuse_b=*/false);
}

__global__ __launch_bounds__(256) void roi_bilinear_wmma(
    const float* __restrict__ img,   // [128,128,1024]
    const int*   __restrict__ rois,  // [512,4] = (x,y,w,h)
    float*       __restrict__ out)   // [512,7,7,1024]
{
    __shared__ float sWx[16][16];                 // B-matrix stage1: [k-slot][sx]
    __shared__ float sWy[16][16];                 // B-matrix stage2: [r-slot][sy]
    __shared__ int   sColOff[NSLOT];              // col-slot -> col*C
    __shared__ int   sRowOff[NSLOT];              // row-slot -> row*W*C
    __shared__ float sH[WAVES][NSLOT][POOL][HPAD];// per-wave x-interp result

    const int roi  = blockIdx.x;
    const int tid  = threadIdx.x;
    const int lane = tid & 31;
    const int wave = tid >> 5;
    const int half = lane >> 4;     // A/B K-pair selector
    const int l15  = lane & 15;     // A: M (channel); B: N (sample)

    // ---- zero weight matrices (pad rows 14,15 stay exactly 0.0) ----
    ((float*)sWx)[tid] = 0.0f;
    ((float*)sWy)[tid] = 0.0f;
    __syncthreads();

    // ---- per-ROI axis samples: threads 0..6 do x-axis, 16..22 do y-axis ----
    if (tid < POOL) {
        const int x = rois[roi * 4 + 0];
        const int w = rois[roi * 4 + 2];
        const float szf = (float)w;
        float c = ((float)tid + 0.5f) * (szf / (float)POOL) - 0.5f;
        c = fminf(fmaxf(c, 0.0f), fmaxf(szf - 1.0f, 0.0f));
        const int i0 = (int)floorf(c);
        const int i1 = imin(i0 + 1, w - 1);
        const float t = c - (float)i0;
        sColOff[tid]        = (x + i0) * C_IMG;
        sColOff[tid + POOL] = (x + i1) * C_IMG;
        sWx[tid][tid]        = 1.0f - t;
        sWx[tid + POOL][tid] = t;
    } else if (tid >= 16 && tid < 16 + POOL) {
        const int s = tid - 16;
        const int y = rois[roi * 4 + 1];
        const int h = rois[roi * 4 + 3];
        const float szf = (float)h;
        float c = ((float)s + 0.5f) * (szf / (float)POOL) - 0.5f;
        c = fminf(fmaxf(c, 0.0f), fmaxf(szf - 1.0f, 0.0f));
        const int i0 = (int)floorf(c);
        const int i1 = imin(i0 + 1, h - 1);
        const float t = c - (float)i0;
        sRowOff[s]        = (y + i0) * (W_IMG * C_IMG);
        sRowOff[s + POOL] = (y + i1) * (W_IMG * C_IMG);
        sWy[s][s]        = 1.0f - t;
        sWy[s + POOL][s] = t;
    }
    __syncthreads();

    // ---- preload B operands (shared across all channel tiles) ----
    // K-chunk kk covers k = 4*kk .. 4*kk+3; this lane holds rows ka, ka+1.
    v2f bx[4], by[4];
#pragma unroll
    for (int kk = 0; kk < 4; ++kk) {
        const int ka = kk * 4 + half * 2;
        bx[kk].x = sWx[ka][l15];     bx[kk].y = sWx[ka + 1][l15];
        by[kk].x = sWy[ka][l15];     by[kk].y = sWy[ka + 1][l15];
    }

    float* const outR = out + (size_t)roi * (POOL * POOL * C_IMG);

    // ---- each wave owns channel tiles ct = wave, wave+8, ... ----
    for (int ct = wave; ct < CTILES; ct += WAVES) {
        const int c0 = ct * 16;
        const int ch = c0 + l15;  // A-matrix M index = channel

        // ===== stage 1: x-interp. H[r]^T[ch, sx] = img[ch, k] x Wx[k, sx] =====
        for (int r = 0; r < NSLOT; ++r) {
            const float* rowp = img + sRowOff[r] + ch;
            v8f acc = {};
#pragma unroll
            for (int kk = 0; kk < 4; ++kk) {
                const int ka = kk * 4 + half * 2;
                const int kb = ka + 1;
                v2f a;
                // clamp padded K slots to valid finite data (weight is 0 there)
                a.x = rowp[sColOff[imin(ka, NSLOT - 1)]];
                a.y = rowp[sColOff[imin(kb, NSLOT - 1)]];
                acc = wmma_f32_k4(a, bx[kk], acc);
            }
            // D elem j in this lane = H[r][sx=l15][chLocal = j + 8*half]
            if (l15 < POOL) {
                float* hp = &sH[wave][r][l15][8 * half];
#pragma unroll
                for (int j = 0; j < 8; ++j) hp[j] = acc[j];
            }
        }

        // ===== stage 2: y-interp. out^T[ch, sy] = H_sx[ch, r] x Wy[r, sy] =====
        for (int sx = 0; sx < POOL; ++sx) {
            v8f acc = {};
#pragma unroll
            for (int kk = 0; kk < 4; ++kk) {
                const int ka = imin(kk * 4 + half * 2, NSLOT - 1);
                const int kb = imin(kk * 4 + half * 2 + 1, NSLOT - 1);
                v2f a;
                a.x = sH[wave][ka][sx][l15];
                a.y = sH[wave][kb][sx][l15];
                acc = wmma_f32_k4(a, by[kk], acc);
            }
            // D elem j = out[sy=l15][sx][c0 + j + 8*half] -> 8 contiguous floats
            if (l15 < POOL) {
                float* op = outR + ((size_t)l15 * POOL + sx) * C_IMG + c0 + 8 * half;
                v4f lo = { acc[0], acc[1], acc[2], acc[3] };
                v4f hi = { acc[4], acc[5], acc[6], acc[7] };
                *(v4f*)op       = lo;
                *((v4f*)op + 1) = hi;
            }
        }
    }
}

extern "C" void kernel_launch(void* const* d_in, const int* in_sizes, int n_in,
                              void* d_out, int out_size, void* d_ws, size_t ws_size,
                              hipStream_t stream) {
    (void)in_sizes; (void)n_in; (void)d_ws; (void)ws_size; (void)out_size;
    const float* img  = (const float*)d_in[0];
    const int*   rois = (const int*)d_in[1];
    float*       out  = (float*)d_out;
    roi_bilinear_wmma<<<dim3(NROI), dim3(256), 0, stream>>>(img, rois, out);
}